// EMDLoss_6262062318052
// MI455X (gfx1250) — compile-verified
//
#include <hip/hip_runtime.h>
#include <math.h>

// EMD (Sinkhorn) for B x N x 3 point clouds, N=2048, eps=0.005, 50 iters.
// Cost tiles are regenerated on the fly via V_WMMA_F32_16X16X4_F32
// (D = P_tile(16x4) x Q_tile(4x16), K padded with 0). Column-side B fragments
// are pre-staged in LDS as (x,y,z,0) quads so each lane grabs its fragment
// with a single aligned ds_load_b64 (no divergent conditional loads in the
// hot loop). Online logsumexp runs in the base-2 logit domain so every exp
// is a bare v_exp_f32.

#define N_PTS 2048
#define EPS_F 0.005f
#define SINK_ITERS 50
#define BLOCK 256
#define NWAVES 8
#define ROWS_PB 16
#define LOG_MARG (-11.090354888959125f)   /* -ln(65536) */
#define LN2_F    (0.6931471805599453f)
// SCALE = log2(e) / eps ; logits are built directly in base-2 domain.
#define SCALE_F  (288.53900817779268f)
#define SCALE2_F (577.07801635558536f)    /* 2*SCALE */

typedef __attribute__((ext_vector_type(2))) float v2f;
typedef __attribute__((ext_vector_type(8))) float v8f;

__device__ __forceinline__ float fast_exp2(float x) {
#if __has_builtin(__builtin_amdgcn_exp2f)
    return __builtin_amdgcn_exp2f(x);   // v_exp_f32 (HW is base-2)
#else
    return exp2f(x);
#endif
}

// merge two (max, sum) pairs in base-2 domain
__device__ __forceinline__ void merge_ms(float& m, float& s, float mo, float so) {
    float mn = fmaxf(m, mo);
    s = s * fast_exp2(m - mn) + so * fast_exp2(mo - mn);
    m = mn;
}

// One half-step: outDual_i = eps*(log(1/65536) - LSE_j((colDual_j - C_ij)/eps))
__global__ __launch_bounds__(BLOCK) void sinkhorn_half(
    const float* __restrict__ rowPts, const float* __restrict__ colPts,
    const float* __restrict__ colDual, float* __restrict__ outDual)
{
    __shared__ float sB[4 * N_PTS];          // staged B frags: (x,y,z,0) per col
    __shared__ float sW[N_PTS];              // (colDual_j - |q_j|^2) * SCALE
    __shared__ float sM[NWAVES][ROWS_PB];
    __shared__ float sS[NWAVES][ROWS_PB];

    const int b = blockIdx.y;
    const int rowbase = blockIdx.x * ROWS_PB;
    const int tid = threadIdx.x;
    const int lane = tid & 31;
    const int wave = tid >> 5;
    const int hi = lane >> 4;     // 0: K=0,1 half / rows v ; 1: K=2,3 half / rows v+8
    const int ln = lane & 15;     // column within tile / row index for A

    // single-pass preload: stage fragments + scaled column constants
    const float* cp = colPts + (size_t)b * 3 * N_PTS;
    const float* cd = colDual + (size_t)b * N_PTS;
    for (int j = tid; j < N_PTS; j += BLOCK) {
        float x = cp[3 * j], y = cp[3 * j + 1], z = cp[3 * j + 2];
        sB[4 * j + 0] = x; sB[4 * j + 1] = y;
        sB[4 * j + 2] = z; sB[4 * j + 3] = 0.0f;
        sW[j] = (cd[j] - (x * x + y * y + z * z)) * SCALE_F;
    }

    // A fragment: 16x4 f32 layout -> lanes 0-15 hold (K0,K1)=(x,y) of row ln,
    // lanes 16-31 hold (K2,K3)=(z,0) of row ln.
    const float* rp = rowPts + ((size_t)b * N_PTS + rowbase) * 3;
    v2f afrag;
    afrag.x = hi ? rp[3 * ln + 2] : rp[3 * ln + 0];
    afrag.y = hi ? 0.0f           : rp[3 * ln + 1];
    float rs[8];                              // |p_i|^2 * SCALE (pre-scaled)
#pragma unroll
    for (int v = 0; v < 8; ++v) {
        int m = v + 8 * hi;
        float x = rp[3 * m], y = rp[3 * m + 1], z = rp[3 * m + 2];
        rs[v] = (x * x + y * y + z * z) * SCALE_F;
    }
    __syncthreads();

    float mrun[8], srun[8];
#pragma unroll
    for (int v = 0; v < 8; ++v) { mrun[v] = -__builtin_inff(); srun[v] = 0.0f; }

    const v2f* sBv = (const v2f*)sB;          // frag (j,hi) at index 2*j+hi
    const int jspan = N_PTS / NWAVES;         // 256 columns per wave, uniform
    const int jstart = wave * jspan;
    for (int jb = jstart; jb < jstart + jspan; jb += 16) {
        int j = jb + ln;
        v2f bfrag = sBv[2 * j + hi];          // one ds_load_b64, no divergence
        v8f c = {};
        v8f d = __builtin_amdgcn_wmma_f32_16x16x4_f32(
            false, afrag, false, bfrag, (short)0, c, false, false);
        float ws = sW[j];
#pragma unroll
        for (int v = 0; v < 8; ++v) {
            // base-2 logit = (colDual_j - C_ij)/eps * log2e = ws - rs + d*2*SCALE
            float logit = fmaf(d[v], SCALE2_F, ws - rs[v]);
            float mo = mrun[v];
            float mn = fmaxf(mo, logit);
            srun[v] = srun[v] * fast_exp2(mo - mn) + fast_exp2(logit - mn);
            mrun[v] = mn;
        }
    }
    // reduce the 16 lanes of each half (they partition this wave's columns)
#pragma unroll
    for (int mask = 1; mask <= 8; mask <<= 1) {
#pragma unroll
        for (int v = 0; v < 8; ++v) {
            float mo = __shfl_xor(mrun[v], mask, 32);
            float so = __shfl_xor(srun[v], mask, 32);
            merge_ms(mrun[v], srun[v], mo, so);
        }
    }
    if (ln == 0) {
#pragma unroll
        for (int v = 0; v < 8; ++v) {
            sM[wave][v + 8 * hi] = mrun[v];
            sS[wave][v + 8 * hi] = srun[v];
        }
    }
    __syncthreads();
    if (tid < ROWS_PB) {
        float m = sM[0][tid], s = sS[0][tid];
        for (int wv = 1; wv < NWAVES; ++wv) merge_ms(m, s, sM[wv][tid], sS[wv][tid]);
        // LSE (natural log) = (m2 + log2(s)) * ln2
        outDual[(size_t)b * N_PTS + rowbase + tid] =
            EPS_F * (LOG_MARG - (m + log2f(s)) * LN2_F);
    }
}

// Per-16-row-block partial of sum_j exp((f+g-C)/eps)*C  (deterministic, no atomics)
__global__ __launch_bounds__(BLOCK) void emd_rowsum(
    const float* __restrict__ preds, const float* __restrict__ gts,
    const float* __restrict__ f, const float* __restrict__ g,
    float* __restrict__ partials)
{
    __shared__ float sB[4 * N_PTS];
    __shared__ float sWQ[N_PTS];   // (g_j - |q_j|^2) * SCALE
    __shared__ float sQ2[N_PTS];   // |q_j|^2
    __shared__ float sAcc[NWAVES][ROWS_PB];

    const int b = blockIdx.y;
    const int rowbase = blockIdx.x * ROWS_PB;
    const int tid = threadIdx.x;
    const int lane = tid & 31;
    const int wave = tid >> 5;
    const int hi = lane >> 4;
    const int ln = lane & 15;

    const float* cp = gts + (size_t)b * 3 * N_PTS;
    const float* gd = g + (size_t)b * N_PTS;
    for (int j = tid; j < N_PTS; j += BLOCK) {
        float x = cp[3 * j], y = cp[3 * j + 1], z = cp[3 * j + 2];
        float q2 = x * x + y * y + z * z;
        sB[4 * j + 0] = x; sB[4 * j + 1] = y;
        sB[4 * j + 2] = z; sB[4 * j + 3] = 0.0f;
        sQ2[j] = q2;
        sWQ[j] = (gd[j] - q2) * SCALE_F;
    }

    const float* rp = preds + ((size_t)b * N_PTS + rowbase) * 3;
    const float* fd = f + (size_t)b * N_PTS + rowbase;
    v2f afrag;
    afrag.x = hi ? rp[3 * ln + 2] : rp[3 * ln + 0];
    afrag.y = hi ? 0.0f           : rp[3 * ln + 1];
    float r[8], frs[8];
#pragma unroll
    for (int v = 0; v < 8; ++v) {
        int m = v + 8 * hi;
        float x = rp[3 * m], y = rp[3 * m + 1], z = rp[3 * m + 2];
        r[v] = x * x + y * y + z * z;
        frs[v] = (fd[m] - r[v]) * SCALE_F;   // (f_i - |p_i|^2) * SCALE
    }
    __syncthreads();

    float acc[8];
#pragma unroll
    for (int v = 0; v < 8; ++v) acc[v] = 0.0f;

    const v2f* sBv = (const v2f*)sB;
    const int jspan = N_PTS / NWAVES;
    const int jstart = wave * jspan;
    for (int jb = jstart; jb < jstart + jspan; jb += 16) {
        int j = jb + ln;
        v2f bfrag = sBv[2 * j + hi];
        v8f c = {};
        v8f d = __builtin_amdgcn_wmma_f32_16x16x4_f32(
            false, afrag, false, bfrag, (short)0, c, false, false);
        float wqs = sWQ[j], q2 = sQ2[j];
#pragma unroll
        for (int v = 0; v < 8; ++v) {
            float Cij = fmaf(d[v], -2.0f, r[v] + q2);          // squared distance
            float arg = fmaf(d[v], SCALE2_F, frs[v] + wqs);    // (f+g-C)/eps * log2e
            acc[v] += fast_exp2(arg) * Cij;
        }
    }
#pragma unroll
    for (int mask = 1; mask <= 8; mask <<= 1) {
#pragma unroll
        for (int v = 0; v < 8; ++v) acc[v] += __shfl_xor(acc[v], mask, 32);
    }
    if (ln == 0) {
#pragma unroll
        for (int v = 0; v < 8; ++v) sAcc[wave][v + 8 * hi] = acc[v];
    }
    __syncthreads();
    if (tid == 0) {
        float t = 0.0f;
        for (int wv = 0; wv < NWAVES; ++wv)
            for (int m = 0; m < ROWS_PB; ++m) t += sAcc[wv][m];
        partials[blockIdx.y * gridDim.x + blockIdx.x] = t;
    }
}

__global__ void zero_kernel(float* p, int n) {
    int i = blockIdx.x * blockDim.x + threadIdx.x;
    if (i < n) p[i] = 0.0f;
}

__global__ void final_reduce(const float* __restrict__ partials, int np,
                             float* __restrict__ out, float Bf) {
    __shared__ float buf[256];
    int per = (np + 255) / 256;
    float s = 0.0f;
    for (int k = 0; k < per; ++k) {
        int idx = (int)threadIdx.x * per + k;
        if (idx < np) s += partials[idx];
    }
    buf[threadIdx.x] = s;
    __syncthreads();
    if (threadIdx.x == 0) {
        float t = 0.0f;
        for (int i = 0; i < 256; ++i) t += buf[i];
        // mean(n * rowsum) over B*N rows == (sum of rowsums) / B
        out[0] = t / Bf;
    }
}

extern "C" void kernel_launch(void* const* d_in, const int* in_sizes, int n_in,
                              void* d_out, int out_size, void* d_ws, size_t ws_size,
                              hipStream_t stream) {
    (void)n_in; (void)out_size; (void)ws_size;
    const float* preds = (const float*)d_in[0];
    const float* gts   = (const float*)d_in[1];
    const int B = in_sizes[0] / (3 * N_PTS);

    float* f = (float*)d_ws;                       // [B, N]
    float* g = f + (size_t)B * N_PTS;              // [B, N]
    float* partials = g + (size_t)B * N_PTS;       // [B * N/16]

    const int zn = 2 * B * N_PTS;                  // zero f and g (contiguous)
    zero_kernel<<<(zn + 255) / 256, 256, 0, stream>>>(f, zn);

    dim3 grid(N_PTS / ROWS_PB, B);
    for (int it = 0; it < SINK_ITERS; ++it) {
        // f-update with old g, then g-update with new f (matches reference order)
        sinkhorn_half<<<grid, BLOCK, 0, stream>>>(preds, gts, g, f);
        sinkhorn_half<<<grid, BLOCK, 0, stream>>>(gts, preds, f, g);
    }
    emd_rowsum<<<grid, BLOCK, 0, stream>>>(preds, gts, f, g, partials);
    final_reduce<<<1, 256, 0, stream>>>(partials, B * (N_PTS / ROWS_PB),
                                        (float*)d_out, (float)B);
}